// History_56083682951394
// MI455X (gfx1250) — compile-verified
//
#include <hip/hip_runtime.h>
#include <hip/hip_bf16.h>
#include <math.h>

#define N_NODES 2000000
#define DDIM    128
#define BROWS   262144
#define CCAP    262144
#define TSTALE  3
#define KPOINT  (BROWS / 2)   // int(B * 0.5), 1-indexed kth smallest

typedef __attribute__((ext_vector_type(2))) float v2f;
typedef __attribute__((ext_vector_type(8))) float v8f;
typedef unsigned int u32;

// ---------------------------------------------------------------------------
// 1) Eviction: for each minibatch gid, clear pos/time and free its slot.
//    gids are unique -> pos/time writes race-free. itg writes are all -1
//    (benign races). reset_idx == -1 wraps to slot C-1 (torch semantics).
// ---------------------------------------------------------------------------
__global__ void evict_kernel(const int* __restrict__ gids,
                             int* __restrict__ pos_ws,
                             int* __restrict__ time_ws,
                             int* __restrict__ itg_ws) {
    int i = blockIdx.x * blockDim.x + threadIdx.x;
    if (i >= BROWS) return;
    int g = gids[i];
    int r = pos_ws[g];
    pos_ws[g]  = -1;
    time_ws[g] = 0;
    int idx = (r < 0) ? (r + CCAP) : r;        // -1 wraps to C-1
    if (idx >= 0 && idx < CCAP) itg_ws[idx] = -1;
}

// ---------------------------------------------------------------------------
// 2) Row norms via V_WMMA_F32_16X16X4_F32.
//    One wave handles 16 rows. A(16x4) = squared grad chunk, B(4x16) = ones,
//    so every column of the f32 accumulator holds the row sum (exact fp32).
//    A layout: lanes 0-15 -> K=k+0,k+1 ; lanes 16-31 -> K=k+2,k+3 (M = lane&15).
//    C/D layout: VGPR j holds M=j (lanes 0-15) and M=j+8 (lanes 16-31).
// ---------------------------------------------------------------------------
__global__ void gnorm_wmma_kernel(const float* __restrict__ grad,
                                  float* __restrict__ gnorm) {
    int gt   = blockIdx.x * blockDim.x + threadIdx.x;
    int wave = gt >> 5;
    int lane = threadIdx.x & 31;
    int row0 = wave * 16;                       // grid sized exactly: no guard
    int m    = lane & 15;
    int kg   = (lane >> 4) << 1;                // 0 or 2

    const float* base = grad + (size_t)(row0 + m) * DDIM + kg;

    v8f c = {};
    v2f bones; bones.x = 1.0f; bones.y = 1.0f;

#pragma unroll
    for (int k = 0; k < DDIM; k += 4) {
        float2 g = *(const float2*)(base + k);
        v2f a; a.x = g.x * g.x; a.y = g.y * g.y;
        // (neg_a, A, neg_b, B, c_mod, C, reuse_a, reuse_b)
        c = __builtin_amdgcn_wmma_f32_16x16x4_f32(false, a, false, bones,
                                                  (short)0, c, false, false);
    }
    // lanes 0..7 write rows 0..7 (c[li]); lanes 16..23 write rows 8..15 (c[li])
    int lg = lane >> 4;
    int li = lane & 15;
    if (li < 8) {
        float s = c[li];
        gnorm[row0 + lg * 8 + li] = sqrtf(s);
    }
}

// ---------------------------------------------------------------------------
// 3) Radix select (kth smallest). Norms are >= 0 so raw f32 bits are
//    monotonic as u32. state: [0]=prefix [1]=k-remaining [2]=thresh(float)
//    [3]=num_free [4]=num_selected
// ---------------------------------------------------------------------------
__global__ void init_state_kernel(u32* state) {
    state[0] = 0u;
    state[1] = (u32)KPOINT;
}

__global__ void zero_hist_kernel(u32* hist) { hist[threadIdx.x] = 0u; }

__global__ void hist_kernel(const float* __restrict__ gn, u32* __restrict__ hist,
                            const u32* __restrict__ state, int shift) {
    __shared__ u32 lh[256];
    lh[threadIdx.x] = 0u;
    __syncthreads();
    u32 prefix = state[0];
    u32 pmask  = (shift == 24) ? 0u : (0xFFFFFFFFu << (shift + 8));
    int stride = gridDim.x * blockDim.x;
    for (int i = blockIdx.x * blockDim.x + threadIdx.x; i < BROWS; i += stride) {
        u32 u = __float_as_uint(gn[i]);
        if ((u & pmask) == prefix) atomicAdd(&lh[(u >> shift) & 255u], 1u);
    }
    __syncthreads();
    atomicAdd(&hist[threadIdx.x], lh[threadIdx.x]);
}

__global__ void select_bin_kernel(const u32* __restrict__ hist, u32* state, int shift) {
    u32 k = state[1], cum = 0u, b = 255u;
    for (u32 i = 0; i < 256u; ++i) {
        u32 h = hist[i];
        if (cum + h >= k) { b = i; break; }
        cum += h;
    }
    state[0] |= (b << shift);
    state[1]  = k - cum;
    if (shift == 0) ((float*)state)[2] = __uint_as_float(state[0]);
}

// ---------------------------------------------------------------------------
// 4) Exclusive scans: mode 0 -> flag = (gnorm <= thresh), mode 1 -> (itg==-1)
// ---------------------------------------------------------------------------
__global__ void scan_block_kernel(const float* __restrict__ gn,
                                  const int* __restrict__ itg,
                                  u32* __restrict__ rank, u32* __restrict__ bsums,
                                  const u32* __restrict__ state, int mode) {
    __shared__ u32 s[256];
    int i = blockIdx.x * 256 + threadIdx.x;
    u32 flag;
    if (mode == 0) {
        float th = ((const float*)state)[2];
        flag = (gn[i] <= th) ? 1u : 0u;
    } else {
        flag = (itg[i] == -1) ? 1u : 0u;
    }
    s[threadIdx.x] = flag;
    __syncthreads();
    for (int off = 1; off < 256; off <<= 1) {
        u32 t = (threadIdx.x >= off) ? s[threadIdx.x - off] : 0u;
        __syncthreads();
        s[threadIdx.x] += t;
        __syncthreads();
    }
    rank[i] = s[threadIdx.x] - flag;            // exclusive in-block
    if (threadIdx.x == 255) bsums[blockIdx.x] = s[255];
}

__global__ void scan_sums_kernel(u32* __restrict__ bsums, u32* __restrict__ total_out) {
    __shared__ u32 s[1024];
    u32 orig = bsums[threadIdx.x];
    s[threadIdx.x] = orig;
    __syncthreads();
    for (int off = 1; off < 1024; off <<= 1) {
        u32 t = (threadIdx.x >= off) ? s[threadIdx.x - off] : 0u;
        __syncthreads();
        s[threadIdx.x] += t;
        __syncthreads();
    }
    bsums[threadIdx.x] = s[threadIdx.x] - orig; // exclusive block offset
    if (threadIdx.x == 1023) *total_out = s[1023];
}

__global__ void add_offsets_kernel(u32* __restrict__ rank, const u32* __restrict__ bsums) {
    int i = blockIdx.x * 256 + threadIdx.x;
    rank[i] += bsums[i >> 8];
}

__global__ void free_scatter_kernel(const int* __restrict__ itg,
                                    const u32* __restrict__ free_rank,
                                    u32* __restrict__ free_list) {
    int s = blockIdx.x * blockDim.x + threadIdx.x;
    if (s >= CCAP) return;
    if (itg[s] == -1) free_list[free_rank[s]] = (u32)s;
}

// ---------------------------------------------------------------------------
// 5) Insert: one 128-thread block per minibatch row.
// ---------------------------------------------------------------------------
__global__ void insert_kernel(const float* __restrict__ feats,
                              const int* __restrict__ gids,
                              const float* __restrict__ gn,
                              const u32* __restrict__ sel_rank,
                              const u32* __restrict__ free_list,
                              int* __restrict__ pos_ws,
                              int* __restrict__ itg_ws,
                              float* __restrict__ emb_out,
                              const u32* __restrict__ state) {
    int i = blockIdx.x;
    float th = ((const float*)state)[2];
    if (!(gn[i] <= th)) return;                 // not selected
    u32 r     = sel_rank[i];
    u32 nfree = state[3];
    if (r >= nfree) {                           // free list exhausted: pad slot == C
        if (threadIdx.x == 0) pos_ws[gids[i]] = CCAP;   // itg/emb scatters dropped
        return;
    }
    int slot = (int)free_list[r];
    emb_out[(size_t)slot * DDIM + threadIdx.x] = feats[(size_t)i * DDIM + threadIdx.x];
    if (threadIdx.x == 0) {
        int g = gids[i];
        pos_ws[g]    = slot;
        itg_ws[slot] = g;
    }
}

// ---------------------------------------------------------------------------
// 6) Age + expire over all N nodes.
// ---------------------------------------------------------------------------
__global__ void age_expire_kernel(int* __restrict__ pos_ws,
                                  int* __restrict__ time_ws,
                                  int* __restrict__ itg_ws) {
    int g = blockIdx.x * blockDim.x + threadIdx.x;
    if (g >= N_NODES) return;
    int p = pos_ws[g];
    int t = time_ws[g] + ((p != -1) ? 1 : 0);
    if (t >= TSTALE) {
        int idx = (p < 0) ? (p + CCAP) : p;     // negative wraps like torch
        if (idx >= 0 && idx < CCAP) itg_ws[idx] = -1;
        p = -1; t = 0;
    }
    pos_ws[g]  = p;
    time_ws[g] = t;
}

// 7) In-place int -> float conversion of the state regions of d_out.
__global__ void finalize_kernel(float* __restrict__ out_pos,
                                float* __restrict__ out_time,
                                float* __restrict__ out_itg) {
    int i = blockIdx.x * blockDim.x + threadIdx.x;
    if (i < N_NODES) {
        int v = ((const int*)out_pos)[i];  out_pos[i]  = (float)v;
        int w = ((const int*)out_time)[i]; out_time[i] = (float)w;
    }
    if (i < CCAP) {
        int v = ((const int*)out_itg)[i];  out_itg[i]  = (float)v;
    }
}

// ---------------------------------------------------------------------------
extern "C" void kernel_launch(void* const* d_in, const int* in_sizes, int n_in,
                              void* d_out, int out_size, void* d_ws, size_t ws_size,
                              hipStream_t stream) {
    // inputs: 0=pos(N i32) 1=time(N i32) 2=emb(C*D f32) 3=itg(C i32)
    //         4=gids(B i32) 5=feats(B*D f32) 6=grad(B*D f32)
    const int*   in_pos  = (const int*)d_in[0];
    const int*   in_time = (const int*)d_in[1];
    const float* in_emb  = (const float*)d_in[2];
    const int*   in_itg  = (const int*)d_in[3];
    const int*   gids    = (const int*)d_in[4];
    const float* feats   = (const float*)d_in[5];
    const float* grad    = (const float*)d_in[6];

    // output layout: emb[C*D] | pos[N] | time[N] | itg[C]  (all float at end;
    // pos/time/itg regions used as int32 scratch during compute)
    float* out     = (float*)d_out;
    float* emb_out = out;
    int*   pos_ws  = (int*)(out + (size_t)CCAP * DDIM);
    int*   time_ws = pos_ws + N_NODES;
    int*   itg_ws  = time_ws + N_NODES;

    // workspace layout
    char* w = (char*)d_ws;
    float* gnorm     = (float*)w;                               w += (size_t)BROWS * 4;
    u32*   sel_rank  = (u32*)w;                                 w += (size_t)BROWS * 4;
    u32*   free_rank = (u32*)w;                                 w += (size_t)CCAP * 4;
    u32*   free_list = (u32*)w;                                 w += (size_t)CCAP * 4;
    u32*   bsumsA    = (u32*)w;                                 w += 1024 * 4;
    u32*   bsumsB    = (u32*)w;                                 w += 1024 * 4;
    u32*   hist      = (u32*)w;                                 w += 256 * 4;
    u32*   state     = (u32*)w;

    // 0) copy initial state into output / scratch views
    hipMemcpyAsync(emb_out, in_emb, sizeof(float) * (size_t)CCAP * DDIM,
                   hipMemcpyDeviceToDevice, stream);
    hipMemcpyAsync(pos_ws,  in_pos,  sizeof(int) * (size_t)N_NODES,
                   hipMemcpyDeviceToDevice, stream);
    hipMemcpyAsync(time_ws, in_time, sizeof(int) * (size_t)N_NODES,
                   hipMemcpyDeviceToDevice, stream);
    hipMemcpyAsync(itg_ws,  in_itg,  sizeof(int) * (size_t)CCAP,
                   hipMemcpyDeviceToDevice, stream);

    // 1) eviction
    evict_kernel<<<BROWS / 256, 256, 0, stream>>>(gids, pos_ws, time_ws, itg_ws);

    // 2) row norms via WMMA (16 rows per wave, exact fp32)
    {
        int waves  = BROWS / 16;            // 16384
        int blocks = (waves * 32) / 256;    // 2048
        gnorm_wmma_kernel<<<blocks, 256, 0, stream>>>(grad, gnorm);
    }

    // 3) kth-smallest threshold by 4-pass radix select
    init_state_kernel<<<1, 1, 0, stream>>>(state);
    for (int pass = 0; pass < 4; ++pass) {
        int shift = 24 - 8 * pass;
        zero_hist_kernel<<<1, 256, 0, stream>>>(hist);
        hist_kernel<<<256, 256, 0, stream>>>(gnorm, hist, state, shift);
        select_bin_kernel<<<1, 1, 0, stream>>>(hist, state, shift);
    }

    // 4) exclusive scan of selection mask (B elems, 1024 blocks of 256)
    scan_block_kernel<<<1024, 256, 0, stream>>>(gnorm, itg_ws, sel_rank, bsumsA, state, 0);
    scan_sums_kernel<<<1, 1024, 0, stream>>>(bsumsA, &state[4]);
    add_offsets_kernel<<<1024, 256, 0, stream>>>(sel_rank, bsumsA);

    //    exclusive scan of free slots (post-eviction itg)
    scan_block_kernel<<<1024, 256, 0, stream>>>(gnorm, itg_ws, free_rank, bsumsB, state, 1);
    scan_sums_kernel<<<1, 1024, 0, stream>>>(bsumsB, &state[3]);
    add_offsets_kernel<<<1024, 256, 0, stream>>>(free_rank, bsumsB);

    free_scatter_kernel<<<CCAP / 256, 256, 0, stream>>>(itg_ws, free_rank, free_list);

    // 5) scatter-insert selected rows into free slots
    insert_kernel<<<BROWS, 128, 0, stream>>>(feats, gids, gnorm, sel_rank, free_list,
                                             pos_ws, itg_ws, emb_out, state);

    // 6) age + expire
    age_expire_kernel<<<(N_NODES + 255) / 256, 256, 0, stream>>>(pos_ws, time_ws, itg_ws);

    // 7) int -> float finalize of pos/time/itg regions
    finalize_kernel<<<(N_NODES + 255) / 256, 256, 0, stream>>>(
        (float*)pos_ws, (float*)time_ws, (float*)itg_ws);
}